// GCN_SumPooling_OneLowRankLayer_18691697672405
// MI455X (gfx1250) — compile-verified
//
#include <hip/hip_runtime.h>

typedef __attribute__((ext_vector_type(2))) float v2f;
typedef __attribute__((ext_vector_type(8))) float v8f;

#define N_NODES 50000
#define N_EDGES 800000
#define D_FEAT 96
#define DIM_H 128
#define N_CLASSES 10

// workspace layout in floats (all offsets 128B-aligned)
#define P_OFF   0        // p[50000]
#define S_OFF   50048    // s[50000]
#define ACC_OFF 100096   // acc[0]=sum relu(s), acc[1]=sum relu(-s)

__global__ __launch_bounds__(256) void zero_buf(float* buf, int n) {
    int i = blockIdx.x * blockDim.x + threadIdx.x;
    if (i < n) buf[i] = 0.0f;
}

// p[i] = dot(x[i,:], a1) via V_WMMA_F32_16X16X4_F32.
// A (16x4) = a1 chunk replicated across the 16 M-rows:
//   lane layout (documented): lanes 0-15 hold K=0,1 ; lanes 16-31 hold K=2,3.
// B (4x16) = x^T chunk: VGPR v, lane n(0-15) holds B[K=v][n], lanes 16-31 hold B[K=v+2][n].
//   => each lane needs x[node][k + 2*hi], x[node][k + 2*hi + 1]: one aligned 8B load.
// D[m][n] = sum_K a1[k+K] * x[n0+n][k+K]  (same value for every m).
__global__ __launch_bounds__(256) void node_proj_wmma(const float* __restrict__ x,
                                                      const float* __restrict__ a1,
                                                      float* __restrict__ p) {
    const int lane = threadIdx.x & 31;
    const int wavesPerBlock = blockDim.x >> 5;
    const int waveId = blockIdx.x * wavesPerBlock + (threadIdx.x >> 5);
    const int nWaves = gridDim.x * wavesPerBlock;
    const int row16 = lane & 15;
    const int koff = (lane >> 4) << 1;          // 0 for lanes 0-15, 2 for lanes 16-31
    const int nTiles = N_NODES / 16;            // 3125 exactly

    for (int t = waveId; t < nTiles; t += nWaves) {
        const int n0 = t << 4;
        const float* xrow = x + (size_t)(n0 + row16) * D_FEAT + koff;
        v8f acc = {};
#pragma unroll
        for (int k = 0; k < D_FEAT; k += 4) {
            v2f bv = *(const v2f*)(xrow + k);            // x[node][k+koff .. +1]
            v2f av = *(const v2f*)(a1 + k + koff);       // a1[k+koff .. +1]
            acc = __builtin_amdgcn_wmma_f32_16x16x4_f32(
                false, av, false, bv, (short)0, acc, false, false);
        }
        // C/D layout: VGPR0, lanes 0-15 -> M=0, N=lane  => p[n0+lane]
        if (lane < 16) p[n0 + lane] = acc[0];
    }
}

__global__ __launch_bounds__(256) void edge_scatter(const int* __restrict__ ei,
                                                    const float* __restrict__ p,
                                                    float* __restrict__ s) {
    int e = blockIdx.x * blockDim.x + threadIdx.x;
    int stride = gridDim.x * blockDim.x;
    for (; e < N_EDGES; e += stride) {
        int src = ei[e];
        int dst = ei[N_EDGES + e];
        atomicAdd(&s[dst], p[src]);   // global_atomic_add_f32, no return
    }
}

__global__ __launch_bounds__(256) void relu_reduce(const float* __restrict__ s,
                                                   float* __restrict__ acc) {
    __shared__ float sp[256];
    __shared__ float sn[256];
    int tid = threadIdx.x;
    int i = blockIdx.x * blockDim.x + tid;
    int stride = gridDim.x * blockDim.x;
    float lp = 0.f, ln = 0.f;
    for (; i < N_NODES; i += stride) {
        float v = s[i];
        lp += fmaxf(v, 0.f);
        ln += fmaxf(-v, 0.f);
    }
    sp[tid] = lp;
    sn[tid] = ln;
    __syncthreads();
    for (int o = 128; o > 0; o >>= 1) {
        if (tid < o) { sp[tid] += sp[tid + o]; sn[tid] += sn[tid + o]; }
        __syncthreads();
    }
    if (tid == 0) {
        atomicAdd(&acc[0], sp[0]);
        atomicAdd(&acc[1], sn[0]);
    }
}

// hG[j] = b1[j]>=0 ? b1[j]*P : (-b1[j])*Nn ;  out[c] = (hG . a2) * b2[c]
__global__ __launch_bounds__(128) void finalize(const float* __restrict__ acc,
                                                const float* __restrict__ b1,
                                                const float* __restrict__ a2,
                                                const float* __restrict__ b2,
                                                float* __restrict__ out) {
    __shared__ float sd[DIM_H];
    int j = threadIdx.x;                 // 128 threads
    float P = acc[0], Nn = acc[1];
    float bj = b1[j];
    float hj = (bj >= 0.f) ? bj * P : (-bj) * Nn;
    sd[j] = hj * a2[j];
    __syncthreads();
    for (int o = 64; o > 0; o >>= 1) {
        if (j < o) sd[j] += sd[j + o];
        __syncthreads();
    }
    if (j < N_CLASSES) out[j] = sd[0] * b2[j];
}

extern "C" void kernel_launch(void* const* d_in, const int* in_sizes, int n_in,
                              void* d_out, int out_size, void* d_ws, size_t ws_size,
                              hipStream_t stream) {
    const float* x  = (const float*)d_in[0];
    const int*   ei = (const int*)d_in[1];     // jax default x64-off => int32
    const float* a1 = (const float*)d_in[2];
    const float* b1 = (const float*)d_in[3];
    const float* a2 = (const float*)d_in[4];
    const float* b2 = (const float*)d_in[5];
    float* ws  = (float*)d_ws;
    float* p   = ws + P_OFF;
    float* s   = ws + S_OFF;
    float* acc = ws + ACC_OFF;
    float* out = (float*)d_out;

    // zero s[50000] and the 2 floats after it (acc is not in this range)
    zero_buf<<<(N_NODES + 2 + 255) / 256, 256, 0, stream>>>(s, N_NODES + 2);

    // p = x @ a1  (WMMA f32 GEMV), 3125 tiles of 16 nodes, 8 waves/block
    node_proj_wmma<<<391, 256, 0, stream>>>(x, a1, p);

    // s[dst] += p[src] over 800000 edges
    edge_scatter<<<(N_EDGES + 255) / 256, 256, 0, stream>>>(ei, p, s);

    // P = sum relu(s), Nn = sum relu(-s)
    relu_reduce<<<196, 256, 0, stream>>>(s, acc);

    // out[10]
    finalize<<<1, DIM_H, 0, stream>>>(acc, b1, a2, b2, out);
}